// PolarOut_44057774522752
// MI455X (gfx1250) — compile-verified
//
#include <hip/hip_runtime.h>
#include <math.h>

typedef __attribute__((ext_vector_type(2))) float v2f;
typedef __attribute__((ext_vector_type(8))) float v8f;

#define LDST 132            // padded dword stride for transposed weights (conflict-free)
#define EPSF 1e-12f
#define S128 0.08838834764831845f   // 128^-0.5
#define S64  0.125f                 // 64^-0.5
#define S32  0.17677669529663687f   // 32^-0.5
#define S16  0.25f                  // 16^-0.5

__device__ __forceinline__ float sigmoidf_(float x) { return 1.0f / (1.0f + __expf(-x)); }
__device__ __forceinline__ float siluf_(float x)    { return x * sigmoidf_(x); }

__global__ void __launch_bounds__(256) polar_zero_kernel(float* __restrict__ p, int n) {
    int i = blockIdx.x * 256 + threadIdx.x;
    if (i < n) p[i] = 0.0f;
}

__global__ void __launch_bounds__(256) polar_node_kernel(
    const float* __restrict__ node_scalar,  // N x 128
    const float* __restrict__ node_equi,    // N x 480
    const int*   __restrict__ batch,        // N
    const float* __restrict__ W_eq0,        // 128 x 64
    const float* __restrict__ b_eq0,        // 64
    const float* __restrict__ W_eq2,        // 32 x 16
    const float* __restrict__ W_out0,       // 64 x 1
    const float* __restrict__ b_out0,       // 1
    const float* __restrict__ W_out2,       // 16
    const float* __restrict__ Ws1,          // 128 x 64
    const float* __restrict__ bs1,          // 64
    const float* __restrict__ Ws2,          // 64 x 2
    const float* __restrict__ bs2,          // 2
    float* __restrict__ out_acc,            // G x 9 (slots 0..5 = polar accumulator)
    int N)
{
    __shared__ __align__(16) float ldsWeq0[64 * LDST];   // transposed: [col][k]
    __shared__ __align__(16) float ldsWs1 [64 * LDST];
    __shared__ __align__(16) float scratch[256 * 3];     // per-node o0, s0, s1

    const int tid  = threadIdx.x;
    const int lane = tid & 31;
    const int wave = tid >> 5;
    const int l15  = lane & 15;
    const int half = lane >> 4;                  // 0 or 1
    const long wgBase = (long)blockIdx.x * 256;

    // ---- stage both 128x64 weight matrices into LDS, transposed ----
    for (int idx = tid; idx < 128 * 64; idx += 256) {
        int k = idx >> 6, c = idx & 63;
        ldsWeq0[c * LDST + k] = W_eq0[idx];
        ldsWs1 [c * LDST + k] = Ws1[idx];
    }
    __syncthreads();

    const float bo0  = b_out0[0];
    const float bs20 = bs2[0];
    const float bs21 = bs2[1];

    // ================= WMMA phase: each wave does 2 tiles of 16 nodes =================
    #pragma unroll 1
    for (int tt = 0; tt < 2; ++tt) {
        const int  tileLocal = wave + tt * 8;                 // 0..15
        const long rowBase   = wgBase + (long)tileLocal * 16;
        long arow = rowBase + l15;
        if (arow >= N) arow = N - 1;                          // clamp (keeps EXEC all-1s)
        const float* aS = node_scalar + arow * 128;           // A for s1 GEMM
        const float* aE = node_equi   + arow * 480;           // A for h0 GEMM (x0 cols 0..127)

        v8f accH[4], accS[4];
        #pragma unroll
        for (int t = 0; t < 4; ++t) {
            accH[t] = (v8f){0.f,0.f,0.f,0.f,0.f,0.f,0.f,0.f};
            accS[t] = (v8f){0.f,0.f,0.f,0.f,0.f,0.f,0.f,0.f};
        }

        #pragma unroll 4
        for (int kk = 0; kk < 32; ++kk) {
            const int acol = kk * 4 + half * 2;               // f32 16x16x4 A layout
            v2f a_h = *(const v2f*)(aE + acol);               // global_load_b64
            v2f a_s = *(const v2f*)(aS + acol);
            const int brow = acol;                            // B layout mirrors A
            #pragma unroll
            for (int t = 0; t < 4; ++t) {
                const int bcol = t * 16 + l15;
                v2f bh = *(const v2f*)(&ldsWeq0[bcol * LDST + brow]);  // ds_load_b64
                v2f bs = *(const v2f*)(&ldsWs1 [bcol * LDST + brow]);
                accH[t] = __builtin_amdgcn_wmma_f32_16x16x4_f32(
                    false, a_h, false, bh, (short)0, accH[t], false, false);
                accS[t] = __builtin_amdgcn_wmma_f32_16x16x4_f32(
                    false, a_s, false, bs, (short)0, accS[t], false, false);
            }
        }

        // ---- epilogue on fragments: silu gates + N=1/N=2 contractions ----
        float po0[8], ps0[8], ps1[8];
        #pragma unroll
        for (int r = 0; r < 8; ++r) { po0[r] = 0.f; ps0[r] = 0.f; ps1[r] = 0.f; }

        #pragma unroll
        for (int t = 0; t < 4; ++t) {
            const int col  = t * 16 + l15;
            const float w0  = W_out0[col];
            const float be  = b_eq0[col];
            const float bsc = bs1[col];
            const float w20 = Ws2[col * 2 + 0];
            const float w21 = Ws2[col * 2 + 1];
            #pragma unroll
            for (int r = 0; r < 8; ++r) {
                float h   = accH[t][r] * S128 + be;           // h0
                float rad = __builtin_sqrtf(h * h + EPSF);
                float g   = h * siluf_(rad);                  // g0
                po0[r] += g * w0;
                float u  = accS[t][r] + bsc;                  // node_scalar@Ws1 + bs1 (no scale)
                float su = siluf_(u);
                ps0[r] += su * w20;
                ps1[r] += su * w21;
            }
        }

        // 16-lane reductions (rows r in lanes 0-15, rows r+8 in lanes 16-31)
        #pragma unroll
        for (int r = 0; r < 8; ++r) {
            float a0 = po0[r], a1 = ps0[r], a2 = ps1[r];
            #pragma unroll
            for (int m = 1; m < 16; m <<= 1) {
                a0 += __shfl_xor(a0, m, 32);
                a1 += __shfl_xor(a1, m, 32);
                a2 += __shfl_xor(a2, m, 32);
            }
            if (l15 == 0) {
                int nodeLocal = tileLocal * 16 + r + half * 8;
                scratch[nodeLocal * 3 + 0] = a0 * S64 + bo0;  // o0
                scratch[nodeLocal * 3 + 1] = a1 + bs20;       // s[:,0]
                scratch[nodeLocal * 3 + 2] = a2 + bs21;       // s[:,1]
            }
        }
    }
    __syncthreads();

    // ================= per-lane phase: h2 / o2 + atomic segment-sum =================
    const long n  = wgBase + tid;
    const bool ok = (n < N);
    const long nc = ok ? n : (long)(N - 1);
    const float* x2p = node_equi + nc * 480 + 320;            // (32, 5) tensor, contiguous

    float acc2[16][5];
    #pragma unroll
    for (int k = 0; k < 16; ++k)
        #pragma unroll
        for (int i = 0; i < 5; ++i) acc2[k][i] = 0.f;

    #pragma unroll 1
    for (int c = 0; c < 8; ++c) {                             // 4 m-rows (20 floats) per chunk
        float buf[20];
        const float4* p4 = (const float4*)(x2p + c * 20);
        #pragma unroll
        for (int q = 0; q < 5; ++q) {
            float4 v = p4[q];
            buf[q * 4 + 0] = v.x; buf[q * 4 + 1] = v.y;
            buf[q * 4 + 2] = v.z; buf[q * 4 + 3] = v.w;
        }
        #pragma unroll
        for (int mm = 0; mm < 4; ++mm) {
            const int m = c * 4 + mm;
            const float* wr = W_eq2 + m * 16;                 // uniform -> scalar loads
            #pragma unroll
            for (int k = 0; k < 16; ++k) {
                float w = wr[k];
                #pragma unroll
                for (int i = 0; i < 5; ++i) acc2[k][i] += buf[mm * 5 + i] * w;
            }
        }
    }

    float o2[5] = {0.f, 0.f, 0.f, 0.f, 0.f};
    #pragma unroll
    for (int k = 0; k < 16; ++k) {
        float h[5]; float ss = EPSF;
        #pragma unroll
        for (int i = 0; i < 5; ++i) { h[i] = acc2[k][i] * S32; ss += h[i] * h[i]; }
        float nrm = __builtin_sqrtf(ss);
        float sil = siluf_(nrm);
        float wo  = W_out2[k];                                // uniform -> scalar load
        #pragma unroll
        for (int i = 0; i < 5; ++i) o2[i] += h[i] * sil * wo;
    }

    if (ok) {
        const float o0v = scratch[tid * 3 + 0];
        const float s0v = scratch[tid * 3 + 1];
        const float s1v = scratch[tid * 3 + 2];
        const int   b   = batch[n];
        float* dst = out_acc + (long)b * 9;
        atomicAdd(&dst[0], o0v * s0v);
        #pragma unroll
        for (int i = 0; i < 5; ++i)
            atomicAdd(&dst[1 + i], o2[i] * S16 * s1v);
    }
}

__global__ void __launch_bounds__(256) polar_finalize_kernel(float* __restrict__ out, int G) {
    int g = blockIdx.x * 256 + threadIdx.x;
    if (g >= G) return;
    float* p = out + (long)g * 9;
    float zero = p[0], dxy = p[1], dyz = p[2], dz2 = p[3], dzx = p[4], dx2y2 = p[5];
    float dn = __builtin_sqrtf(dxy*dxy + dyz*dyz + dz2*dz2 + dzx*dzx + dx2y2*dx2y2);
    const float c = 0.5773502691896258f;   // 3^-0.5
    float xx = c * (dn - dz2) + dx2y2 + zero;
    float yy = c * (dn - dz2) - dx2y2 + zero;
    float zz = c * (dn + 2.0f * dz2) + zero;
    p[0] = xx;  p[1] = dxy; p[2] = dzx;
    p[3] = dxy; p[4] = yy;  p[5] = dyz;
    p[6] = dzx; p[7] = dyz; p[8] = zz;
}

extern "C" void kernel_launch(void* const* d_in, const int* in_sizes, int n_in,
                              void* d_out, int out_size, void* d_ws, size_t ws_size,
                              hipStream_t stream) {
    (void)n_in; (void)d_ws; (void)ws_size;
    const float* node_scalar = (const float*)d_in[0];
    const float* node_equi   = (const float*)d_in[1];
    const int*   batch       = (const int*)d_in[2];
    const float* W_eq0  = (const float*)d_in[3];
    const float* b_eq0  = (const float*)d_in[4];
    const float* W_eq2  = (const float*)d_in[5];
    const float* W_out0 = (const float*)d_in[6];
    const float* b_out0 = (const float*)d_in[7];
    const float* W_out2 = (const float*)d_in[8];
    const float* Ws1    = (const float*)d_in[9];
    const float* bs1    = (const float*)d_in[10];
    const float* Ws2    = (const float*)d_in[11];
    const float* bs2    = (const float*)d_in[12];
    float* out = (float*)d_out;

    const int N = in_sizes[0] / 128;   // number of nodes
    const int G = out_size / 9;        // number of graphs

    polar_zero_kernel<<<(out_size + 255) / 256, 256, 0, stream>>>(out, out_size);
    polar_node_kernel<<<(N + 255) / 256, 256, 0, stream>>>(
        node_scalar, node_equi, batch,
        W_eq0, b_eq0, W_eq2, W_out0, b_out0, W_out2,
        Ws1, bs1, Ws2, bs2, out, N);
    polar_finalize_kernel<<<(G + 255) / 256, 256, 0, stream>>>(out, G);
}